// Encoder_39874476376345
// MI455X (gfx1250) — compile-verified
//
#include <hip/hip_runtime.h>

typedef __attribute__((ext_vector_type(2))) float v2f;
typedef __attribute__((ext_vector_type(8))) float v8f;

#define FEAT1 256
#define HID   1600
#define OUTF  400
#define OUTP  448   // W2/t2 padded to 7*64 columns so GEMM2 needs no guards

// ---------- degree / normalization ----------
__global__ void k_init_deg(float* __restrict__ deg, int n) {
  int i = blockIdx.x * blockDim.x + threadIdx.x;
  if (i < n) deg[i] = 1.0f;  // self-loop contributes 1
}

__global__ void k_count_deg(const int* __restrict__ dst, float* __restrict__ deg, int E) {
  int e = blockIdx.x * blockDim.x + threadIdx.x;
  if (e < E) atomicAdd(&deg[dst[e]], 1.0f);
}

__global__ void k_dinv(const float* __restrict__ deg, float* __restrict__ dinv, int n) {
  int i = blockIdx.x * blockDim.x + threadIdx.x;
  if (i < n) dinv[i] = rsqrtf(deg[i]);  // deg >= 1 always (self-loops)
}

// ---------- layer-1 aggregation on x (256 wide = 64 float4) ----------
__global__ void k_agg1_init(const float4* __restrict__ x4, const float* __restrict__ dinv,
                            float4* __restrict__ agg4, int total) {
  int t = blockIdx.x * blockDim.x + threadIdx.x;
  if (t >= total) return;
  int i = t >> 6;
  float s = dinv[i]; s *= s;            // self-loop weight dinv^2
  float4 v = x4[t];
  v.x *= s; v.y *= s; v.z *= s; v.w *= s;
  agg4[t] = v;
}

__global__ void k_agg1_edges(const int* __restrict__ src, const int* __restrict__ dst,
                             const float* __restrict__ dinv, const float4* __restrict__ x4,
                             float* __restrict__ agg, int total) {
  int t = blockIdx.x * blockDim.x + threadIdx.x;
  if (t >= total) return;
  int e = t >> 6, c = t & 63;
  int s = src[e], d = dst[e];
  float w = dinv[s] * dinv[d];
  float4 v = x4[(size_t)s * 64 + c];
  float* o = agg + (size_t)d * FEAT1 + c * 4;
  atomicAdd(o + 0, v.x * w);
  atomicAdd(o + 1, v.y * w);
  atomicAdd(o + 2, v.z * w);
  atomicAdd(o + 3, v.w * w);
}

// ---------- weight reformat into WMMA-fragment-major layout ----------
// Bf[k/4][n][k%4]  -> a B fragment (two K values for this lane's half) is one
// aligned b64 load; a wave's fragment load is one coalesced 256B burst.
__global__ void k_fmt_w1(const float* __restrict__ W, float* __restrict__ Wf, int total) {
  int t = blockIdx.x * blockDim.x + threadIdx.x;
  if (t >= total) return;
  int k = t / HID, n = t % HID;
  Wf[((size_t)(k >> 2) * HID + n) * 4 + (k & 3)] = W[(size_t)k * HID + n];
}

__global__ void k_fmt_w2(const float* __restrict__ W, float* __restrict__ Wf, int total) {
  int t = blockIdx.x * blockDim.x + threadIdx.x;
  if (t >= total) return;
  int k = t / OUTP, n = t % OUTP;
  float v = (n < OUTF) ? W[(size_t)k * OUTF + n] : 0.0f;
  Wf[((size_t)(k >> 2) * OUTP + n) * 4 + (k & 3)] = v;
}

// ---------- f32 WMMA GEMM: C[M,N] = A[M,K] @ B[K,N] (+bias, +relu) ----------
// One wave -> 32x64 output strip: 2 A fragments x 4 B fragments -> 8 WMMAs per
// K-step from 6 b64 loads. B is in fragment-major layout (see k_fmt_*).
// Grids sized so there are no guards: EXEC is all-1s at every WMMA.
__global__ void k_gemm_wmma(const float* __restrict__ A, const float* __restrict__ Bf,
                            const float* __restrict__ bias, float* __restrict__ C,
                            int N, int K, int relu) {
  int wave = threadIdx.x >> 5;
  int lane = threadIdx.x & 31;
  int wpb  = blockDim.x >> 5;
  int n0 = (blockIdx.y * wpb + wave) * 64;
  int m0 = blockIdx.x * 32;
  int mi = lane & 15;          // M (A) or N (B/C) position within tile
  int kk = (lane >> 4) << 1;   // 0 for lanes 0-15, 2 for lanes 16-31

  v8f acc00 = {}, acc01 = {}, acc02 = {}, acc03 = {};
  v8f acc10 = {}, acc11 = {}, acc12 = {}, acc13 = {};

  const float* arow0 = A + (size_t)(m0 + mi) * K + kk;
  const float* arow1 = arow0 + (size_t)16 * K;
  const float* bbase = Bf + ((size_t)(n0 + mi)) * 4 + kk;

#pragma unroll 2
  for (int k = 0; k < K; k += 4) {
    v2f a0 = *(const v2f*)(arow0 + k);
    v2f a1 = *(const v2f*)(arow1 + k);
    const float* bb = bbase + (size_t)(k >> 2) * N * 4;
    v2f b0 = *(const v2f*)(bb);
    v2f b1 = *(const v2f*)(bb + 64);
    v2f b2 = *(const v2f*)(bb + 128);
    v2f b3 = *(const v2f*)(bb + 192);
    acc00 = __builtin_amdgcn_wmma_f32_16x16x4_f32(false, a0, false, b0, (short)0, acc00, false, false);
    acc01 = __builtin_amdgcn_wmma_f32_16x16x4_f32(false, a0, false, b1, (short)0, acc01, false, false);
    acc02 = __builtin_amdgcn_wmma_f32_16x16x4_f32(false, a0, false, b2, (short)0, acc02, false, false);
    acc03 = __builtin_amdgcn_wmma_f32_16x16x4_f32(false, a0, false, b3, (short)0, acc03, false, false);
    acc10 = __builtin_amdgcn_wmma_f32_16x16x4_f32(false, a1, false, b0, (short)0, acc10, false, false);
    acc11 = __builtin_amdgcn_wmma_f32_16x16x4_f32(false, a1, false, b1, (short)0, acc11, false, false);
    acc12 = __builtin_amdgcn_wmma_f32_16x16x4_f32(false, a1, false, b2, (short)0, acc12, false, false);
    acc13 = __builtin_amdgcn_wmma_f32_16x16x4_f32(false, a1, false, b3, (short)0, acc13, false, false);
  }

  // C/D layout: VGPR r = row r (lanes 0-15) / row 8+r (lanes 16-31), col = lane&15
  int rhi = (lane >> 4) * 8;
  v8f accs[2][4] = {{acc00, acc01, acc02, acc03}, {acc10, acc11, acc12, acc13}};
#pragma unroll
  for (int mt = 0; mt < 2; ++mt) {
#pragma unroll
    for (int t = 0; t < 4; ++t) {
      int n = n0 + t * 16 + mi;
      float bv = bias ? bias[n] : 0.0f;
#pragma unroll
      for (int r = 0; r < 8; ++r) {
        float v = accs[mt][t][r] + bv;
        if (relu) v = fmaxf(v, 0.0f);
        C[(size_t)(m0 + mt * 16 + rhi + r) * N + n] = v;
      }
    }
  }
}

// ---------- layer-2 aggregation on t2 = h@W2 (400 wide = 100 float4) ----------
__global__ void k_out_init(const float4* __restrict__ t2p4, const float* __restrict__ dinv,
                           const float4* __restrict__ b2_4, float4* __restrict__ out4, int total) {
  int t = blockIdx.x * blockDim.x + threadIdx.x;
  if (t >= total) return;
  int i = t / 100, c = t % 100;
  float s = dinv[i]; s *= s;
  float4 v = t2p4[(size_t)i * 112 + c];   // padded stride 448/4
  float4 bb = b2_4[c];
  float4 o;
  o.x = v.x * s + bb.x;
  o.y = v.y * s + bb.y;
  o.z = v.z * s + bb.z;
  o.w = v.w * s + bb.w;
  out4[t] = o;
}

__global__ void k_out_edges(const int* __restrict__ src, const int* __restrict__ dst,
                            const float* __restrict__ dinv, const float4* __restrict__ t2p4,
                            float* __restrict__ out, int total) {
  int t = blockIdx.x * blockDim.x + threadIdx.x;
  if (t >= total) return;
  int e = t / 100, c = t % 100;
  int s = src[e], d = dst[e];
  float w = dinv[s] * dinv[d];
  float4 v = t2p4[(size_t)s * 112 + c];
  float* o = out + (size_t)d * OUTF + c * 4;
  atomicAdd(o + 0, v.x * w);
  atomicAdd(o + 1, v.y * w);
  atomicAdd(o + 2, v.z * w);
  atomicAdd(o + 3, v.w * w);
}

extern "C" void kernel_launch(void* const* d_in, const int* in_sizes, int n_in,
                              void* d_out, int out_size, void* d_ws, size_t ws_size,
                              hipStream_t stream) {
  const float* x  = (const float*)d_in[0];
  const int*   ei = (const int*)d_in[1];
  const float* W1 = (const float*)d_in[2];
  const float* b1 = (const float*)d_in[3];
  const float* W2 = (const float*)d_in[4];
  const float* b2 = (const float*)d_in[5];

  const int n = in_sizes[0] / FEAT1;  // 20000 (multiple of 32)
  const int E = in_sizes[1] / 2;      // 320000
  const int* src = ei;
  const int* dst = ei + E;

  char* ws = (char*)d_ws;
  size_t off = 0;
  auto walloc = [&](size_t bytes) -> void* {
    void* p = ws + off;
    off += (bytes + 255) & ~(size_t)255;
    return p;
  };
  float* deg  = (float*)walloc((size_t)n * 4);
  float* dinv = (float*)walloc((size_t)n * 4);
  float* agg1 = (float*)walloc((size_t)n * FEAT1 * 4);   // (A x)  [n,256]
  float* h    = (float*)walloc((size_t)n * HID * 4);     // relu((A x)W1 + b1)  [n,1600]
  float* w1f  = (float*)walloc((size_t)FEAT1 * HID * 4); // W1 fragment-major
  float* w2f  = (float*)walloc((size_t)HID * OUTP * 4);  // W2 padded + fragment-major
  float* t2p  = (float*)walloc((size_t)n * OUTP * 4);    // h@W2 padded [n,448]

  const int TB = 256;
  // normalization
  k_init_deg<<<(n + TB - 1) / TB, TB, 0, stream>>>(deg, n);
  k_count_deg<<<(E + TB - 1) / TB, TB, 0, stream>>>(dst, deg, E);
  k_dinv<<<(n + TB - 1) / TB, TB, 0, stream>>>(deg, dinv, n);

  // weight reformat (independent of aggregation; tiny)
  int totf1 = FEAT1 * HID;
  k_fmt_w1<<<(totf1 + TB - 1) / TB, TB, 0, stream>>>(W1, w1f, totf1);
  int totf2 = HID * OUTP;
  k_fmt_w2<<<(totf2 + TB - 1) / TB, TB, 0, stream>>>(W2, w2f, totf2);

  // layer 1: aggregate x first (linearity: A(xW1) == (Ax)W1), then WMMA GEMM
  int tot1i = n * 64;
  k_agg1_init<<<(tot1i + TB - 1) / TB, TB, 0, stream>>>((const float4*)x, dinv,
                                                        (float4*)agg1, tot1i);
  int tot1e = E * 64;
  k_agg1_edges<<<(tot1e + TB - 1) / TB, TB, 0, stream>>>(src, dst, dinv, (const float4*)x,
                                                         agg1, tot1e);
  // GEMM1: [n,256]x[256,1600] -> h ; M tiles of 32 (625), 1600 = 5 waves * 5 blocks * 64
  k_gemm_wmma<<<dim3(n / 32, 5), 160, 0, stream>>>(agg1, w1f, b1, h, HID, FEAT1, 1);

  // layer 2: GEMM first (t2 = h@W2), then aggregate 400-wide
  // GEMM2: [n,1600]x[1600,448] -> t2p ; 448 = 7 waves * 64, exact
  k_gemm_wmma<<<dim3(n / 32, 1), 224, 0, stream>>>(h, w2f, nullptr, t2p, OUTP, HID, 0);

  float* out = (float*)d_out;
  int tot2i = n * 100;
  k_out_init<<<(tot2i + TB - 1) / TB, TB, 0, stream>>>((const float4*)t2p, dinv,
                                                       (const float4*)b2, (float4*)out, tot2i);
  int tot2e = E * 100;
  k_out_edges<<<(tot2e + TB - 1) / TB, TB, 0, stream>>>(src, dst, dinv, (const float4*)t2p,
                                                        out, tot2e);
}